// HeteroSAGE_47090021433544
// MI455X (gfx1250) — compile-verified
//
#include <hip/hip_runtime.h>

// ---------------------------------------------------------------------------
// HeteroSAGE (2-layer, bipartite Q/P graph), D = H = 128, f32.
// GEMMs use V_WMMA_F32_16X16X4_F32; neighbor means via L2 f32 atomics.
// Each GEMM block computes a 128x128 output tile: the 128x128 weight is
// staged into LDS once and reused by 8 row-tiles (256 WMMAs / wave / block).
// ---------------------------------------------------------------------------

typedef __attribute__((ext_vector_type(2))) float v2f;
typedef __attribute__((ext_vector_type(8))) float v8f;

static constexpr int DIM = 128;  // D == H == 128

#define F_MEAN 1   // A = msum / max(deg,1)
#define F_ACC  2   // C += result (else C = result + bias)
#define F_RELU 4   // final relu(C)
#define F_W2   8   // use W0+W1 (and b0+b1)

// ---------------------------- utility kernels ------------------------------

__global__ void fill_zero_f4(float4* __restrict__ p, long long n4) {
  long long i = (long long)blockIdx.x * blockDim.x + threadIdx.x;
  if (i < n4) p[i] = make_float4(0.f, 0.f, 0.f, 0.f);
}

__global__ void fill_zero_f(float* __restrict__ p, long long n) {
  long long i = (long long)blockIdx.x * blockDim.x + threadIdx.x;
  if (i < n) p[i] = 0.f;
}

__global__ void degree_count(const int* __restrict__ dst, int E,
                             float* __restrict__ deg) {
  int e = blockIdx.x * blockDim.x + threadIdx.x;
  if (e < E) atomicAdd(&deg[dst[e]], 1.0f);
}

// One thread per (edge, 4-column group). Gather float4 from source row
// (resident in L2: hQ=51MB, hP=102MB, L2=192MB), scatter-add into msum.
__global__ void scatter_add(const float* __restrict__ hsrc,
                            const int* __restrict__ src,
                            const int* __restrict__ dst,
                            int E, float* __restrict__ msum) {
  long long tid = (long long)blockIdx.x * blockDim.x + threadIdx.x;
  if (tid >= (long long)E * (DIM / 4)) return;
  int e = (int)(tid >> 5);          // 32 groups of 4 columns per edge
  int g = (int)(tid & 31) << 2;     // column base
  int s = src[e];
  int d = dst[e];
  float4 v = *(const float4*)(hsrc + (long long)s * DIM + g);
  float* o = msum + (long long)d * DIM + g;
  atomicAdd(o + 0, v.x);
  atomicAdd(o + 1, v.y);
  atomicAdd(o + 2, v.z);
  atomicAdd(o + 3, v.w);
}

// ------------------------------ WMMA GEMM ----------------------------------
// Block = 256 threads = 8 wave32s. Block computes a 128x128 tile of C:
// weight staged once in LDS, then 8 row-tiles of 16. Per row-tile each wave
// computes one 16x16 tile over K=128 via 32x V_WMMA_F32_16X16X4_F32 on two
// independent accumulator chains.
//
//   C[row0:row0+128, :] (+)= A' @ (W0 (+W1)) (+ b0 (+b1)),  A' = A or msum/deg
//
__global__ __launch_bounds__(256)
void sage_gemm(const float* __restrict__ A,      // [M,128]
               const float* __restrict__ deg,    // [M] (F_MEAN only)
               const float* __restrict__ W0,     // [128,128]
               const float* __restrict__ W1,     // [128,128] or null
               const float* __restrict__ b0,     // [128] or null
               const float* __restrict__ b1,     // [128] or null
               float* __restrict__ C,            // [M,128]
               int M, int flags) {
  __shared__ float sA[128][132];  // padded: conflict-free A-frag reads
  __shared__ float sW[128][136];  // padded: rows k and k+2 hit disjoint banks
  __shared__ float sB[128];       // summed bias
  __shared__ float sS[128];       // per-row 1/max(deg,1) (mean mode)

  const int t = threadIdx.x;
  const int row0 = blockIdx.x * 128;

  // ---- per-row scale + bias (barrier before A staging uses sS) ----
  if (t < 128) {
    float sc = 1.0f;
    if (flags & F_MEAN) {
      int gr = row0 + t;
      float dg = (gr < M) ? deg[gr] : 1.0f;
      sc = 1.0f / fmaxf(dg, 1.0f);
    }
    sS[t] = sc;
    float bv = 0.f;
    if (b0) bv = b0[t] + ((flags & F_W2) ? b1[t] : 0.f);
    sB[t] = bv;
  }
  __syncthreads();

  // ---- stage W (optionally W0+W1): 4096 float4s, 16 per thread ----
  {
    const float4* W0v = (const float4*)W0;
    const float4* W1v = (const float4*)W1;
    const bool two = (flags & F_W2) != 0;
#pragma unroll
    for (int i = 0; i < 16; ++i) {
      int f4 = t + 256 * i;          // 0..4095
      int r = f4 >> 5;               // 32 float4 per 128-col row
      int c = (f4 & 31) << 2;
      float4 w = W0v[f4];
      if (two) {
        float4 w1 = W1v[f4];
        w.x += w1.x; w.y += w1.y; w.z += w1.z; w.w += w1.w;
      }
      sW[r][c + 0] = w.x; sW[r][c + 1] = w.y;
      sW[r][c + 2] = w.z; sW[r][c + 3] = w.w;
    }
  }
  // ---- stage A block (128x128), scaled: 4096 float4s, 16 per thread ----
  {
#pragma unroll
    for (int i = 0; i < 16; ++i) {
      int f4 = t + 256 * i;          // 0..4095
      int r = f4 >> 5;               // local row
      int c = (f4 & 31) << 2;
      int gr = row0 + r;
      float4 v = make_float4(0.f, 0.f, 0.f, 0.f);
      if (gr < M) v = *(const float4*)(A + (long long)gr * DIM + c);
      float sc = sS[r];
      sA[r][c + 0] = v.x * sc; sA[r][c + 1] = v.y * sc;
      sA[r][c + 2] = v.z * sc; sA[r][c + 3] = v.w * sc;
    }
  }
  __syncthreads();

  const int wave = t >> 5;
  const int lane = t & 31;
  const int half = lane >> 4;     // 0: low half, 1: high half per WMMA layout
  const int lm   = lane & 15;
  const int col  = (wave << 4) + lm;
  const float bv = sB[col];

  // 8 row-tiles reuse the staged weight (uniform loop: EXEC stays all-ones
  // across every WMMA; only loads/stores are per-element guarded).
  for (int tt = 0; tt < 8; ++tt) {
    const int arow = (tt << 4) + lm;     // local A row for this wave's frag
    v8f acc0 = {};
    v8f acc1 = {};
    if (!(flags & F_ACC)) {
#pragma unroll
      for (int i = 0; i < 8; ++i) acc0[i] = bv;
    }

    // K = 128 in steps of 8: two independent 16x16x4 WMMA chains.
#pragma unroll
    for (int k = 0; k < 128; k += 8) {
      {
        int ka = k + 2 * half;     // A frag: v0 holds K=k / k+2 per lane-half
        v2f a, b;
        a.x = sA[arow][ka];    a.y = sA[arow][ka + 1];
        b.x = sW[ka][col];     b.y = sW[ka + 1][col];
        acc0 = __builtin_amdgcn_wmma_f32_16x16x4_f32(
            false, a, false, b, (short)0, acc0, false, false);
      }
      {
        int kb = k + 4 + 2 * half;
        v2f a, b;
        a.x = sA[arow][kb];    a.y = sA[arow][kb + 1];
        b.x = sW[kb][col];     b.y = sW[kb + 1][col];
        acc1 = __builtin_amdgcn_wmma_f32_16x16x4_f32(
            false, a, false, b, (short)0, acc1, false, false);
      }
    }

    // C/D layout: VGPR i -> tile row (i + 8*half), column = lane&15.
#pragma unroll
    for (int i = 0; i < 8; ++i) {
      int gr = row0 + (tt << 4) + (half << 3) + i;
      if (gr < M) {
        float v = acc0[i] + acc1[i];
        long long off = (long long)gr * DIM + col;
        if (flags & F_ACC) v += C[off];
        if (flags & F_RELU) v = fmaxf(v, 0.0f);
        C[off] = v;
      }
    }
  }
}

// ------------------------------- host side ---------------------------------

extern "C" void kernel_launch(void* const* d_in, const int* in_sizes, int n_in,
                              void* d_out, int out_size, void* d_ws,
                              size_t ws_size, hipStream_t stream) {
  (void)n_in; (void)out_size; (void)ws_size;

  const float* xq = (const float*)d_in[0];   // [NQ,128]
  const float* xp = (const float*)d_in[1];   // [NP,128]
  const float* Ws = (const float*)d_in[2];   // [2,4,128,128]
  const float* Wn = (const float*)d_in[3];   // [2,4,128,128]
  const float* B  = (const float*)d_in[4];   // [2,4,128]
  const int* click_src  = (const int*)d_in[5];
  const int* click_dst  = (const int*)d_in[6];
  const int* qres_src   = (const int*)d_in[7];
  const int* qres_dst   = (const int*)d_in[8];
  const int* rclick_src = (const int*)d_in[9];
  const int* rclick_dst = (const int*)d_in[10];
  const int* rqres_src  = (const int*)d_in[11];
  const int* rqres_dst  = (const int*)d_in[12];
  const int Ec  = in_sizes[5];
  const int Eq  = in_sizes[7];
  const int Erc = in_sizes[9];
  const int Erq = in_sizes[11];

  const int NQ = in_sizes[0] / DIM;
  const int NP = in_sizes[1] / DIM;

  // Workspace: hQ1 | hP1 | msum(NP*128) | deg(NP)   (~257 MB)
  float* hQ1  = (float*)d_ws;
  float* hP1  = hQ1 + (size_t)NQ * DIM;
  float* msum = hP1 + (size_t)NP * DIM;
  float* deg  = msum + (size_t)NP * DIM;

  float* outQ = (float*)d_out;
  float* outP = outQ + (size_t)NQ * DIM;

  auto W_self = [&](int l, int r) { return Ws + (size_t)(l * 4 + r) * DIM * DIM; };
  auto W_nei  = [&](int l, int r) { return Wn + (size_t)(l * 4 + r) * DIM * DIM; };
  auto bias   = [&](int l, int r) { return B + (size_t)(l * 4 + r) * DIM; };

  auto gemm = [&](const float* A, const float* dg, const float* W0,
                  const float* W1, const float* b0, const float* b1,
                  float* C, int M, int flags) {
    sage_gemm<<<(unsigned)((M + 127) / 128), 256, 0, stream>>>(
        A, dg, W0, W1, b0, b1, C, M, flags);
  };

  // mean over in-neighbors of one relation, then C += mean @ Wn_r
  auto mean_into = [&](const float* h_src_feat, const int* src, const int* dst,
                       int E, int n_dst, const float* Wnr, float* C,
                       int extra_flags) {
    long long nm = (long long)n_dst * DIM;
    fill_zero_f4<<<(unsigned)((nm / 4 + 255) / 256), 256, 0, stream>>>(
        (float4*)msum, nm / 4);
    fill_zero_f<<<(unsigned)((n_dst + 255) / 256), 256, 0, stream>>>(
        deg, n_dst);
    degree_count<<<(unsigned)((E + 255) / 256), 256, 0, stream>>>(dst, E, deg);
    long long tot = (long long)E * (DIM / 4);
    scatter_add<<<(unsigned)((tot + 255) / 256), 256, 0, stream>>>(
        h_src_feat, src, dst, E, msum);
    gemm(msum, deg, Wnr, nullptr, nullptr, nullptr, C, n_dst,
         F_MEAN | F_ACC | extra_flags);
  };

  const float* inQ = xq;
  const float* inP = xp;
  for (int l = 0; l < 2; ++l) {
    float* oQ = (l == 0) ? hQ1 : outQ;
    float* oP = (l == 0) ? hP1 : outP;
    const int relu = (l == 0) ? F_RELU : 0;

    // ---- dst = P : relations 0 (Click Q->P) and 3 (QueryResult Q->P) ----
    // fc_self for both relations folded into one GEMM: hP @ (Ws0 + Ws3) + (b0 + b3)
    gemm(inP, nullptr, W_self(l, 0), W_self(l, 3), bias(l, 0), bias(l, 3),
         oP, NP, F_W2);
    mean_into(inQ, click_src, click_dst, Ec, NP, W_nei(l, 0), oP, 0);
    mean_into(inQ, qres_src, qres_dst, Eq, NP, W_nei(l, 3), oP, relu);

    // ---- dst = Q : relations 1 (rev_Click P->Q) and 2 (rev_QueryResult P->Q)
    gemm(inQ, nullptr, W_self(l, 1), W_self(l, 2), bias(l, 1), bias(l, 2),
         oQ, NQ, F_W2);
    mean_into(inP, rclick_src, rclick_dst, Erc, NQ, W_nei(l, 1), oQ, 0);
    mean_into(inP, rqres_src, rqres_dst, Erq, NQ, W_nei(l, 2), oQ, relu);

    inQ = oQ;
    inP = oP;
  }
}